// NonLocal_71485435674646
// MI455X (gfx1250) — compile-verified
//
#include <hip/hip_runtime.h>
#include <hip/hip_bf16.h>

typedef float v2f __attribute__((ext_vector_type(2)));
typedef float v8f __attribute__((ext_vector_type(8)));

#define Bn 4
#define Cc 64
#define ICc 32
#define Hh 128
#define Ww_ 128
#define Nn (Hh * Ww_)      // 16384
#define MP (Nn / 4)        // 4096 pooled pixels

// ---------------------------------------------------------------------------
// Kernel 1: fused 1x1 conv + 2x2 maxpool.
// x:[B][C][H][W], w:[IC][C], bias:[IC] -> outp:[B][IC][MP]
// Lanes map to consecutive pooled pixels m (coalesced stores, wave-uniform
// weight index -> scalar loads). Each thread does 4 dot products + max.
// ---------------------------------------------------------------------------
__global__ void convpool_kernel(const float* __restrict__ x,
                                const float* __restrict__ w,
                                const float* __restrict__ bias,
                                float* __restrict__ outp) {
  int t = blockIdx.x * blockDim.x + threadIdx.x;   // B*IC*MP threads
  int m = t & (MP - 1);
  int i = (t >> 12) & (ICc - 1);
  int b = t >> 17;
  int mr = m >> 6;          // pooled row  0..63
  int mc = m & 63;          // pooled col  0..63
  const float* xb = x + (size_t)b * Cc * Nn + (size_t)(mr * 2) * Ww_ + mc * 2;
  float a0 = 0.f, a1 = 0.f, a2 = 0.f, a3 = 0.f;
#pragma unroll 8
  for (int c = 0; c < Cc; ++c) {
    float wv = w[i * Cc + c];
    const float* p = xb + (size_t)c * Nn;
    a0 = fmaf(wv, p[0], a0);
    a1 = fmaf(wv, p[1], a1);
    a2 = fmaf(wv, p[Ww_], a2);
    a3 = fmaf(wv, p[Ww_ + 1], a3);
  }
  float r = fmaxf(fmaxf(a0, a1), fmaxf(a2, a3)) + bias[i];
  outp[((size_t)(b * ICc + i) << 12) + m] = r;
}

// ---------------------------------------------------------------------------
// Kernel 2: A[b] = g[b] * phi[b]^T   (32x32 = sum over 4096 pooled pixels)
// One block per batch; LDS-tiled (pad 129 to avoid bank conflicts).
// ---------------------------------------------------------------------------
__global__ void gramian_kernel(const float* __restrict__ phim,
                               const float* __restrict__ gm,
                               float* __restrict__ A) {
  __shared__ float sp[ICc][129];
  __shared__ float sg[ICc][129];
  int b = blockIdx.x;
  int t = threadIdx.x;            // 256 threads
  int i = t >> 3;                 // 0..31  (row of g)
  int j0 = (t & 7) * 4;           // 0..28  (4 rows of phi)
  float acc0 = 0.f, acc1 = 0.f, acc2 = 0.f, acc3 = 0.f;
  for (int m0 = 0; m0 < MP; m0 += 128) {
    __syncthreads();
    for (int k = t; k < ICc * 128; k += 256) {
      int row = k >> 7;
      int col = k & 127;
      size_t src = ((size_t)(b * ICc + row) << 12) + m0 + col;
      sp[row][col] = phim[src];
      sg[row][col] = gm[src];
    }
    __syncthreads();
#pragma unroll 4
    for (int m = 0; m < 128; ++m) {
      float gv = sg[i][m];
      acc0 = fmaf(gv, sp[j0 + 0][m], acc0);
      acc1 = fmaf(gv, sp[j0 + 1][m], acc1);
      acc2 = fmaf(gv, sp[j0 + 2][m], acc2);
      acc3 = fmaf(gv, sp[j0 + 3][m], acc3);
    }
  }
  float* Ab = A + b * ICc * ICc + i * ICc + j0;
  Ab[0] = acc0; Ab[1] = acc1; Ab[2] = acc2; Ab[3] = acc3;
}

// ---------------------------------------------------------------------------
// Kernel 3: fold everything into a per-batch 64x64 matrix + 64 bias:
//   Meff = (1/N) * W_w * A * theta_w + I       beff = (1/N)*W_w*A*theta_b + W_b
// 4 blocks x 64 threads (thread = output row r).
// ---------------------------------------------------------------------------
__global__ void fold_kernel(const float* __restrict__ A,
                            const float* __restrict__ theta_w,
                            const float* __restrict__ theta_b,
                            const float* __restrict__ W_w,
                            const float* __restrict__ W_b,
                            float* __restrict__ Meff,
                            float* __restrict__ beff) {
  int b = blockIdx.x;
  int r = threadIdx.x;                 // 0..63
  const float* Ab = A + b * ICc * ICc;
  const float invN = 1.0f / (float)Nn;
  float tmp[ICc];                      // row r of (W_w * A)  [1x32]
#pragma unroll
  for (int j = 0; j < ICc; ++j) {
    float s = 0.f;
#pragma unroll
    for (int i = 0; i < ICc; ++i) s = fmaf(W_w[r * ICc + i], Ab[i * ICc + j], s);
    tmp[j] = s;
  }
  float* Mr = Meff + b * Cc * Cc + r * Cc;
  for (int c = 0; c < Cc; ++c) {
    float s = 0.f;
#pragma unroll
    for (int j = 0; j < ICc; ++j) s = fmaf(tmp[j], theta_w[j * Cc + c], s);
    Mr[c] = s * invN + ((r == c) ? 1.0f : 0.0f);
  }
  float s2 = 0.f;
#pragma unroll
  for (int j = 0; j < ICc; ++j) s2 = fmaf(tmp[j], theta_b[j], s2);
  beff[b * Cc + r] = s2 * invN + W_b[r];
}

// ---------------------------------------------------------------------------
// Kernel 4: apply: out[b] = Meff[b] (64x64) @ target[b] (64x16384) + beff[b]
// WMMA f32 16x16x4. One wave per 16-pixel column tile; 4 row-tiles (all 64
// output channels) accumulated simultaneously in 4 x v8f. K-loop: 16 chunks.
// A-fragment layout (f32 16x4): lanes 0-15 -> K{0,1}, lanes 16-31 -> K{2,3};
// B-fragment mirrors with N in low lane bits. C/D: VGPR v -> row v + 8*hi.
// ---------------------------------------------------------------------------
__global__ void apply_kernel(const float* __restrict__ X,
                             const float* __restrict__ Meff,
                             const float* __restrict__ beff,
                             float* __restrict__ Y) {
  int wave = (blockIdx.x * blockDim.x + threadIdx.x) >> 5;
  int lane = threadIdx.x & 31;
  int b = wave >> 10;                  // 1024 n-tiles per batch
  int n0 = (wave & 1023) << 4;
  int ml = lane & 15;
  int hi = lane >> 4;
  int k0 = hi << 1;

  const float* Xb = X + (size_t)b * Cc * Nn;
  const float* Mw = Meff + b * Cc * Cc;
  float* Yb = Y + (size_t)b * Cc * Nn;

  v8f acc[4];
#pragma unroll
  for (int mt = 0; mt < 4; ++mt)
    acc[mt] = (v8f){0.f, 0.f, 0.f, 0.f, 0.f, 0.f, 0.f, 0.f};

#pragma unroll
  for (int kk = 0; kk < Cc; kk += 4) {
    v2f bf;
    bf.x = Xb[(size_t)(kk + k0) * Nn + n0 + ml];
    bf.y = Xb[(size_t)(kk + k0 + 1) * Nn + n0 + ml];
#pragma unroll
    for (int mt = 0; mt < 4; ++mt) {
      // aligned float2 load: (kk + k0) is even, rows are 64 floats
      v2f af = *(const v2f*)(Mw + (mt * 16 + ml) * Cc + kk + k0);
      acc[mt] = __builtin_amdgcn_wmma_f32_16x16x4_f32(
          false, af, false, bf, (short)0, acc[mt], false, false);
    }
  }

#pragma unroll
  for (int mt = 0; mt < 4; ++mt) {
#pragma unroll
    for (int v = 0; v < 8; ++v) {
      int row = mt * 16 + hi * 8 + v;
      Yb[(size_t)row * Nn + n0 + ml] = acc[mt][v] + beff[b * Cc + row];
    }
  }
}

// ---------------------------------------------------------------------------
extern "C" void kernel_launch(void* const* d_in, const int* in_sizes, int n_in,
                              void* d_out, int out_size, void* d_ws, size_t ws_size,
                              hipStream_t stream) {
  const float* target    = (const float*)d_in[0];
  const float* ref       = (const float*)d_in[1];
  const float* ref_align = (const float*)d_in[2];
  const float* theta_w   = (const float*)d_in[3];
  const float* theta_b   = (const float*)d_in[4];
  const float* phi_w     = (const float*)d_in[5];
  const float* phi_b     = (const float*)d_in[6];
  const float* g_w       = (const float*)d_in[7];
  const float* g_b       = (const float*)d_in[8];
  const float* W_w       = (const float*)d_in[9];
  const float* W_b       = (const float*)d_in[10];
  float* out = (float*)d_out;

  float* ws   = (float*)d_ws;
  float* phim = ws;                                   // B*IC*MP = 524288
  float* gm   = phim + (size_t)Bn * ICc * MP;         // 524288
  float* Amat = gm + (size_t)Bn * ICc * MP;           // B*32*32 = 4096
  float* Meff = Amat + Bn * ICc * ICc;                // B*64*64 = 16384
  float* beff = Meff + Bn * Cc * Cc;                  // B*64    = 256

  // 1) conv1x1 + maxpool for phi (from ref) and g (from ref_align)
  convpool_kernel<<<(Bn * ICc * MP) / 256, 256, 0, stream>>>(ref, phi_w, phi_b, phim);
  convpool_kernel<<<(Bn * ICc * MP) / 256, 256, 0, stream>>>(ref_align, g_w, g_b, gm);
  // 2) A = g phi^T per batch
  gramian_kernel<<<Bn, 256, 0, stream>>>(phim, gm, Amat);
  // 3) fold into effective 64x64 + bias
  fold_kernel<<<Bn, Cc, 0, stream>>>(Amat, theta_w, theta_b, W_w, W_b, Meff, beff);
  // 4) out = (Meff) @ target + beff   (WMMA)  — 4096 waves
  apply_kernel<<<(Bn * (Nn / 16) * 32) / 256, 256, 0, stream>>>(target, Meff, beff, out);
}